// SchemaDisambiguator_34351148433904
// MI455X (gfx1250) — compile-verified
//
#include <hip/hip_runtime.h>
#include <hip/hip_bf16.h>

// Problem constants (match reference)
#define BB 2
#define NN 8192
#define FIN 512
#define EE 65536
#define HH 8
#define DD 64
#define CC 512   // HH*DD

typedef float v2f __attribute__((ext_vector_type(2)));
typedef float v8f __attribute__((ext_vector_type(8)));

// -------------------------------------------------------------------------
// Kernel 1: h = x @ W.T  via V_WMMA_F32_16X16X4_F32 (exact fp32)
//   X: (B*N, FIN) row-major, W: (CC, FIN) row-major, Hout: (B*N, CC)
//   Block = 128 threads (4 waves). Block handles one 16-row M tile and a
//   256-col strip; each wave register-blocks a 16x64 strip (4 x 16x16 tiles)
//   so each LDS A-fragment feeds 4 WMMAs. X tile staged in LDS (32 KB).
// -------------------------------------------------------------------------
__global__ __launch_bounds__(128) void gat_gemm_wmma(
    const float* __restrict__ X, const float* __restrict__ W,
    float* __restrict__ Hout) {
  const int tilesN = CC / 256;             // 2 blocks along columns
  const int mb = blockIdx.x / tilesN;      // 0..(B*N/16 - 1)
  const int nb = blockIdx.x % tilesN;      // 0..1
  const int m0 = mb * 16;
  const int lane = threadIdx.x & 31;
  const int wave = threadIdx.x >> 5;       // 0..3
  const int c0 = nb * 256 + wave * 64;     // wave's 64-col strip

  __shared__ float lds_x[16 * FIN];        // 32 KB X tile

  // Cooperative float4 load of the 16x512 X tile into LDS
  const float4* xs = (const float4*)(X + (size_t)m0 * FIN);
  float4* xd = (float4*)lds_x;
  #pragma unroll
  for (int i = 0; i < (16 * FIN / 4) / 128; ++i)
    xd[i * 128 + threadIdx.x] = xs[i * 128 + threadIdx.x];
  __syncthreads();

  // Fragment base pointers per the 16x4 A / 4x16 B VGPR layouts:
  //   lanes 0-15 carry K=0,1  ; lanes 16-31 carry K=2,3
  const int half = lane >> 4;              // 0 or 1
  const int l15 = lane & 15;
  const float* arow = lds_x + l15 * FIN + half * 2;            // A: row m0+l15
  const float* brow0 = W + (size_t)(c0 + 0  + l15) * FIN + half * 2;
  const float* brow1 = W + (size_t)(c0 + 16 + l15) * FIN + half * 2;
  const float* brow2 = W + (size_t)(c0 + 32 + l15) * FIN + half * 2;
  const float* brow3 = W + (size_t)(c0 + 48 + l15) * FIN + half * 2;

  // Warm the L2/L1 path for the W strips (emits global_prefetch_b8)
  __builtin_prefetch(brow0, 0, 0);
  __builtin_prefetch(brow1, 0, 0);
  __builtin_prefetch(brow2, 0, 0);
  __builtin_prefetch(brow3, 0, 0);

  v8f acc0 = {}, acc1 = {}, acc2 = {}, acc3 = {};
  #pragma unroll 4
  for (int k = 0; k < FIN; k += 4) {
    v2f afrag = *(const v2f*)(arow + k);    // ds_load_b64, reused 4x
    v2f b0 = *(const v2f*)(brow0 + k);      // global_load_b64 (L2-resident)
    v2f b1 = *(const v2f*)(brow1 + k);
    v2f b2 = *(const v2f*)(brow2 + k);
    v2f b3 = *(const v2f*)(brow3 + k);
    acc0 = __builtin_amdgcn_wmma_f32_16x16x4_f32(false, afrag, false, b0,
                                                 (short)0, acc0, false, false);
    acc1 = __builtin_amdgcn_wmma_f32_16x16x4_f32(false, afrag, false, b1,
                                                 (short)0, acc1, false, false);
    acc2 = __builtin_amdgcn_wmma_f32_16x16x4_f32(false, afrag, false, b2,
                                                 (short)0, acc2, false, false);
    acc3 = __builtin_amdgcn_wmma_f32_16x16x4_f32(false, afrag, false, b3,
                                                 (short)0, acc3, false, false);
  }

  // C/D layout: VGPR v -> (M = v + 8*half, N = l15)
  const int row = m0 + half * 8;
  #pragma unroll
  for (int v = 0; v < 8; ++v) {
    Hout[(size_t)(row + v) * CC + (c0 + 0  + l15)] = acc0[v];
    Hout[(size_t)(row + v) * CC + (c0 + 16 + l15)] = acc1[v];
    Hout[(size_t)(row + v) * CC + (c0 + 32 + l15)] = acc2[v];
    Hout[(size_t)(row + v) * CC + (c0 + 48 + l15)] = acc3[v];
  }
}

// -------------------------------------------------------------------------
// Kernel 2a/2b: last_edge[n] = max{e : src[e]==n}, else -1
// -------------------------------------------------------------------------
__global__ void gat_init_last_edge(int* __restrict__ last_edge) {
  int n = blockIdx.x * blockDim.x + threadIdx.x;
  if (n < NN) last_edge[n] = -1;
}

__global__ void gat_scan_src(const int* __restrict__ src,
                             int* __restrict__ last_edge) {
  int e = blockIdx.x * blockDim.x + threadIdx.x;
  if (e < EE) atomicMax(&last_edge[src[e]], e);
}

// -------------------------------------------------------------------------
// Kernel 3: per-node attention weights.
//   For node n with e = last_edge[n] (note src[e]==n):
//     s_b = leakyrelu( h[b,n,hh]·a1 + h[b,dst[e],hh]·a2 , 0.2)
//     w[b,n,hh] = softmax over b in {0,1}
// -------------------------------------------------------------------------
__global__ void gat_compute_w(const float* __restrict__ Hmat,
                              const int* __restrict__ edge_dst,
                              const int* __restrict__ last_edge,
                              const float* __restrict__ avec,
                              float* __restrict__ Wout) {
  int idx = blockIdx.x * blockDim.x + threadIdx.x;  // N*H
  if (idx >= NN * HH) return;
  int n = idx >> 3;
  int hh = idx & 7;
  int e = last_edge[n];
  float w0 = 0.f, w1 = 0.f;
  if (e >= 0) {
    int dn = edge_dst[e];
    float s[2];
    #pragma unroll
    for (int b = 0; b < 2; ++b) {
      const float* hs = Hmat + ((size_t)(b * NN + n) * CC) + hh * DD;
      const float* hd = Hmat + ((size_t)(b * NN + dn) * CC) + hh * DD;
      float acc = 0.f;
      #pragma unroll 8
      for (int d = 0; d < DD; ++d)
        acc = fmaf(hs[d], avec[d], fmaf(hd[d], avec[DD + d], acc));
      s[b] = acc > 0.f ? acc : 0.2f * acc;
    }
    float mx = fmaxf(s[0], s[1]);
    float e0 = __expf(s[0] - mx), e1 = __expf(s[1] - mx);
    float inv = __frcp_rn(e0 + e1);
    w0 = e0 * inv;
    w1 = e1 * inv;
  }
  Wout[(size_t)(0 * NN + n) * HH + hh] = w0;
  Wout[(size_t)(1 * NN + n) * HH + hh] = w1;
}

// -------------------------------------------------------------------------
// Kernel 4a: zero agg buffer; 4b: scatter-add agg[:,src[e]] += h[:,dst[e]]
// -------------------------------------------------------------------------
__global__ void gat_zero(float* __restrict__ p, int n4) {
  int i = blockIdx.x * blockDim.x + threadIdx.x;
  if (i < n4) ((float4*)p)[i] = make_float4(0.f, 0.f, 0.f, 0.f);
}

__global__ __launch_bounds__(256) void gat_scatter(
    const float* __restrict__ Hmat, const int* __restrict__ esrc,
    const int* __restrict__ edst, float* __restrict__ agg) {
  long long i = (long long)blockIdx.x * 256 + threadIdx.x;  // E * (B*CC)
  if (i >= (long long)EE * (BB * CC)) return;
  int e = (int)(i >> 10);       // / (B*CC) = /1024
  int r = (int)(i & 1023);
  int b = r >> 9;               // / CC
  int c = r & (CC - 1);
  int sn = esrc[e];
  int dn = edst[e];
  float val = Hmat[((size_t)(b * NN + dn)) * CC + c];
  atomicAdd(&agg[((size_t)(b * NN + sn)) * CC + c], val);
}

// -------------------------------------------------------------------------
// Kernel 5: out[b,n,d] = (1/H) * sum_hh w[b,n,hh] * agg[b,n,hh,d]
// -------------------------------------------------------------------------
__global__ void gat_finalize(const float* __restrict__ agg,
                             const float* __restrict__ Wv,
                             float* __restrict__ out) {
  int i = blockIdx.x * blockDim.x + threadIdx.x;  // B*N*D
  if (i >= BB * NN * DD) return;
  int d = i & (DD - 1);
  int n = (i >> 6) & (NN - 1);
  int b = i >> 19;
  const float* ag = agg + (size_t)(b * NN + n) * CC;
  const float* wv = Wv + (size_t)(b * NN + n) * HH;
  float acc = 0.f;
  #pragma unroll
  for (int hh = 0; hh < HH; ++hh)
    acc = fmaf(wv[hh], ag[hh * DD + d], acc);
  out[i] = acc * (1.0f / HH);
}

// -------------------------------------------------------------------------
// Launch
// -------------------------------------------------------------------------
extern "C" void kernel_launch(void* const* d_in, const int* in_sizes, int n_in,
                              void* d_out, int out_size, void* d_ws,
                              size_t ws_size, hipStream_t stream) {
  (void)in_sizes; (void)n_in; (void)out_size; (void)ws_size;
  const float* x = (const float*)d_in[0];          // (B,N,FIN)
  const int* edges = (const int*)d_in[1];          // (2,E)
  const float* W = (const float*)d_in[2];          // (CC,FIN)
  const float* a = (const float*)d_in[3];          // (2*DD,1)
  float* out = (float*)d_out;                      // (B,N,DD)

  const int* esrc = edges;
  const int* edst = edges + EE;

  // Workspace layout
  char* ws = (char*)d_ws;
  float* h_buf = (float*)ws;                       // B*N*CC floats
  ws += (size_t)BB * NN * CC * sizeof(float);
  float* agg = (float*)ws;                         // B*N*CC floats
  ws += (size_t)BB * NN * CC * sizeof(float);
  float* w_buf = (float*)ws;                       // B*N*H floats
  ws += (size_t)BB * NN * HH * sizeof(float);
  int* last_edge = (int*)ws;                       // N ints

  // 1) GEMM: h = x @ W.T   (grid: (B*N/16) M-tiles x 2 column strips)
  {
    dim3 grid((BB * NN / 16) * (CC / 256));
    gat_gemm_wmma<<<grid, 128, 0, stream>>>(x, W, h_buf);
  }

  // 2) last_edge
  gat_init_last_edge<<<(NN + 255) / 256, 256, 0, stream>>>(last_edge);
  gat_scan_src<<<(EE + 255) / 256, 256, 0, stream>>>(esrc, last_edge);

  // 3) attention weights (softmax over batch at the selected edge)
  gat_compute_w<<<(NN * HH + 255) / 256, 256, 0, stream>>>(
      h_buf, edst, last_edge, a, w_buf);

  // 4) zero agg, then scatter-add
  {
    int n4 = BB * NN * CC / 4;
    gat_zero<<<(n4 + 255) / 256, 256, 0, stream>>>(agg, n4);
    long long total = (long long)EE * (BB * CC);
    int blocks = (int)((total + 255) / 256);
    gat_scatter<<<blocks, 256, 0, stream>>>(h_buf, esrc, edst, agg);
  }

  // 5) finalize: weighted mean over heads
  gat_finalize<<<(BB * NN * DD + 255) / 256, 256, 0, stream>>>(agg, w_buf, out);
}